// IPMP_IPA_67070209294572
// MI455X (gfx1250) — compile-verified
//
#include <hip/hip_runtime.h>
#include <hip/hip_bf16.h>
#include <math.h>

typedef __attribute__((ext_vector_type(2))) float v2f;
typedef __attribute__((ext_vector_type(8))) float v8f;

#define BB 2
#define LL 2048
#define NN (BB*LL)      // 4096 nodes
#define KK 32
#define DD 64
#define HH 4
#define CC 64
#define PP 4
#define VPP 8
#define NPROJ 960       // 256 (q) + 512 (kv) + 48 (qp) + 144 (kvp)
#define SDIM 640
#define BK 64           // GEMM K-chunk

// ---------------------------------------------------------------------------
// Pack [Wq | Wkv | Wqp | Wkvp] -> Wcat [64][960], biases likewise
// ---------------------------------------------------------------------------
__global__ void pack_w_kernel(const float* __restrict__ Wq,  const float* __restrict__ bq,
                              const float* __restrict__ Wkv, const float* __restrict__ bkv,
                              const float* __restrict__ Wqp, const float* __restrict__ bqp,
                              const float* __restrict__ Wkvp,const float* __restrict__ bkvp,
                              float* __restrict__ Wcat, float* __restrict__ bcat)
{
    int i = blockIdx.x * blockDim.x + threadIdx.x;
    if (i < DD * NPROJ) {
        int d = i / NPROJ, j = i % NPROJ;
        float v;
        if      (j < 256) v = Wq  [d * 256 + j];
        else if (j < 768) v = Wkv [d * 512 + (j - 256)];
        else if (j < 816) v = Wqp [d * 48  + (j - 768)];
        else              v = Wkvp[d * 144 + (j - 816)];
        Wcat[i] = v;
    }
    if (i < NPROJ) {
        float v;
        if      (i < 256) v = bq[i];
        else if (i < 768) v = bkv[i - 256];
        else if (i < 816) v = bqp[i - 768];
        else              v = bkvp[i - 816];
        bcat[i] = v;
    }
}

// ---------------------------------------------------------------------------
// Rigid frames from backbone: R (cols e1,e2,e3) + t, 12 floats per node
// ---------------------------------------------------------------------------
__global__ void frames_kernel(const float* __restrict__ X, float* __restrict__ F)
{
    int n = blockIdx.x * blockDim.x + threadIdx.x;
    if (n >= NN) return;
    const float* x = X + (size_t)n * 9;
    float Nv[3], CA[3], Cc[3];
#pragma unroll
    for (int i = 0; i < 3; ++i) {
        Nv[i] = x[i]     * 0.1f;   // / POS_SCALE
        CA[i] = x[3 + i] * 0.1f;
        Cc[i] = x[6 + i] * 0.1f;
    }
    float v1[3], v2[3];
#pragma unroll
    for (int i = 0; i < 3; ++i) { v1[i] = Cc[i] - CA[i]; v2[i] = Nv[i] - CA[i]; }
    float n1 = rsqrtf(v1[0]*v1[0] + v1[1]*v1[1] + v1[2]*v1[2]);
    float e1[3];
#pragma unroll
    for (int i = 0; i < 3; ++i) e1[i] = v1[i] * n1;
    float d12 = e1[0]*v2[0] + e1[1]*v2[1] + e1[2]*v2[2];
    float u2[3];
#pragma unroll
    for (int i = 0; i < 3; ++i) u2[i] = v2[i] - e1[i] * d12;
    float n2 = rsqrtf(u2[0]*u2[0] + u2[1]*u2[1] + u2[2]*u2[2]);
    float e2[3];
#pragma unroll
    for (int i = 0; i < 3; ++i) e2[i] = u2[i] * n2;
    float e3[3];
    e3[0] = e1[1]*e2[2] - e1[2]*e2[1];
    e3[1] = e1[2]*e2[0] - e1[0]*e2[2];
    e3[2] = e1[0]*e2[1] - e1[1]*e2[0];
    float* f = F + (size_t)n * 12;
#pragma unroll
    for (int i = 0; i < 3; ++i) {
        f[i*3 + 0] = e1[i];   // R[i][0]
        f[i*3 + 1] = e2[i];   // R[i][1]
        f[i*3 + 2] = e3[i];   // R[i][2]
        f[9 + i]   = CA[i];   // t
    }
}

// ---------------------------------------------------------------------------
// fp32 WMMA GEMM:  Y[M,N] = A[M,Kd] @ W[Kd,N] + bias  (opt. ReLU)
// Block: 128 threads = 4 waves; block tile 64(M) x 64(N); K chunked by 64.
// A staged row-major, W staged TRANSPOSED (Wt[col][k]) so both A and B
// K-pairs are contiguous 8B-aligned LDS loads (single ds_load_b64 each,
// no v_mov repacking). Each A fragment feeds 4 v_wmma_f32_16x16x4_f32.
// Requires: M % 64 == 0, N % 64 == 0, Kd % 64 == 0.
// ---------------------------------------------------------------------------
__global__ void __launch_bounds__(128)
gemm_kernel(const float* __restrict__ A, const float* __restrict__ W,
            const float* __restrict__ bias, float* __restrict__ Y,
            int M, int N, int Kd, int relu)
{
    __shared__ float As[64][BK + 4];
    __shared__ float Wt[64][BK + 4];         // Wt[col][k]
    const int m0  = blockIdx.x * 64;
    const int n0  = blockIdx.y * 64;
    const int tid = threadIdx.x;
    const int wave = tid >> 5;
    const int lane = tid & 31;
    const int half = lane >> 4;              // 0: lanes 0-15, 1: lanes 16-31
    const int ar   = lane & 15;

    v8f acc[4];
#pragma unroll
    for (int j = 0; j < 4; ++j)
        acc[j] = (v8f){0.f, 0.f, 0.f, 0.f, 0.f, 0.f, 0.f, 0.f};

    for (int k0 = 0; k0 < Kd; k0 += BK) {
        // cooperative staging: A coalesced row-major, W coalesced read +
        // transposed LDS write
        for (int i = tid; i < 64 * BK; i += 128) {
            int r = i >> 6, c = i & 63;
            As[r][c] = A[(size_t)(m0 + r) * Kd + k0 + c];
        }
        for (int i = tid; i < BK * 64; i += 128) {
            int r = i >> 6, c = i & 63;
            Wt[c][r] = W[(size_t)(k0 + r) * N + n0 + c];
        }
        __syncthreads();
#pragma unroll
        for (int kk = 0; kk < BK; kk += 4) {
            v2f av = *(const v2f*)&As[wave * 16 + ar][kk + 2 * half];
#pragma unroll
            for (int j = 0; j < 4; ++j) {
                v2f bv = *(const v2f*)&Wt[j * 16 + ar][kk + 2 * half];
                acc[j] = __builtin_amdgcn_wmma_f32_16x16x4_f32(
                             false, av, false, bv, (short)0, acc[j], false, false);
            }
        }
        __syncthreads();
    }

#pragma unroll
    for (int j = 0; j < 4; ++j) {
        const int col = n0 + j * 16 + ar;
        const float bz = bias[col];
#pragma unroll
        for (int r = 0; r < 8; ++r) {
            float v = acc[j][r] + bz;
            if (relu) v = fmaxf(v, 0.f);
            Y[(size_t)(m0 + wave * 16 + r + 8 * half) * N + col] = v;
        }
    }
}

// ---------------------------------------------------------------------------
// Apply frames to projected points: q_pts [n][16][3], kv_pts [n][48][3]
// lin layout (per node, from Ylin): qp at col 768 (48), kvp at col 816 (144);
// coordinate j of point i is lin[j*npts + i] (jnp.split semantics).
// ---------------------------------------------------------------------------
__global__ void frame_points_kernel(const float* __restrict__ Ylin,
                                    const float* __restrict__ F,
                                    float* __restrict__ qp, float* __restrict__ kvp)
{
    int n = blockIdx.x * blockDim.x + threadIdx.x;
    if (n >= NN) return;
    const float* f = F + (size_t)n * 12;
    float R[9], t3[3];
#pragma unroll
    for (int i = 0; i < 9; ++i) R[i] = f[i];
#pragma unroll
    for (int i = 0; i < 3; ++i) t3[i] = f[9 + i];

    const float* lin = Ylin + (size_t)n * NPROJ + 768;
    float* q = qp + (size_t)n * 48;
    for (int i = 0; i < 16; ++i) {
        float px = lin[i], py = lin[16 + i], pz = lin[32 + i];
        q[i*3+0] = R[0]*px + R[1]*py + R[2]*pz + t3[0];
        q[i*3+1] = R[3]*px + R[4]*py + R[5]*pz + t3[1];
        q[i*3+2] = R[6]*px + R[7]*py + R[8]*pz + t3[2];
    }
    const float* lin2 = Ylin + (size_t)n * NPROJ + 816;
    float* kv = kvp + (size_t)n * 144;
    for (int i = 0; i < 48; ++i) {
        float px = lin2[i], py = lin2[48 + i], pz = lin2[96 + i];
        kv[i*3+0] = R[0]*px + R[1]*py + R[2]*pz + t3[0];
        kv[i*3+1] = R[3]*px + R[4]*py + R[5]*pz + t3[1];
        kv[i*3+2] = R[6]*px + R[7]*py + R[8]*pz + t3[2];
    }
}

// ---------------------------------------------------------------------------
// Attention: one block (128 thr) per node; 128 = K(32) x H(4) logits.
// Produces s[node][640] = [o(256) | opt.x(32) | opt.y(32) | opt.z(32) |
//                          opt_norm(32) | o_pair(256)]
// ---------------------------------------------------------------------------
__global__ void __launch_bounds__(128)
attn_kernel(const float* __restrict__ Ylin, const float* __restrict__ qpts,
            const float* __restrict__ kvpts, const float* __restrict__ F,
            const float* __restrict__ hE, const int* __restrict__ Eidx,
            const float* __restrict__ maskA, const float* __restrict__ Wb,
            const float* __restrict__ bb, const float* __restrict__ hwraw,
            float* __restrict__ S)
{
    const int n   = blockIdx.x;
    const int bI  = n >> 11;                 // n / L
    const int tid = threadIdx.x;

    __shared__ float q_s[256];
    __shared__ float qp_s[48];
    __shared__ float Rt_s[12];
    __shared__ int   eix[KK];
    __shared__ float logit[128];
    __shared__ float aw[128];
    __shared__ float red[8];
    __shared__ float opt_raw[96];

    q_s[tid]       = Ylin[(size_t)n * NPROJ + tid];
    q_s[tid + 128] = Ylin[(size_t)n * NPROJ + tid + 128];
    if (tid < 48) qp_s[tid] = qpts[(size_t)n * 48 + tid];
    if (tid < 12) Rt_s[tid] = F[(size_t)n * 12 + tid];
    if (tid < KK) eix[tid]  = bI * LL + Eidx[(size_t)n * KK + tid];
    __syncthreads();

    const int k = tid >> 2;                  // neighbor slot
    const int h = tid & 3;                   // head
    const int e = eix[k];

    // warm the lines phase 2 will stream (v rows and v_pts rows of neighbors)
    __builtin_prefetch(Ylin + (size_t)e * NPROJ + 256 + 64 + h * 128, 0, 1);
    __builtin_prefetch(kvpts + (size_t)e * 144 + 12 + h * 36, 0, 1);

    // q . k_gathered (C=64)
    const float* krow = Ylin + (size_t)e * NPROJ + 256 + h * 128;
    const float* qrow = q_s + h * 64;
    float dot = 0.f;
#pragma unroll 8
    for (int c = 0; c < CC; ++c) dot += qrow[c] * krow[c];

    // b = h_E @ Wb
    const float* he = hE + ((size_t)n * KK + k) * DD;
    float bt = bb[h];
#pragma unroll 8
    for (int c = 0; c < DD; ++c) bt += he[c] * Wb[c * HH + h];

    // point term
    float hw = log1pf(expf(hwraw[h]));       // softplus
    const float* kp = kvpts + (size_t)e * 144 + h * 36;
    const float* qp = qp_s + h * 12;
    float pts = 0.f;
#pragma unroll
    for (int p = 0; p < PP; ++p) {
        float dx = qp[p*3+0] - kp[p*3+0];
        float dy = qp[p*3+1] - kp[p*3+1];
        float dz = qp[p*3+2] - kp[p*3+2];
        pts += dx*dx + dy*dy + dz*dz;
    }

    const float SC_QK = 0.07216878364870323f;   // sqrt(1/(3C)), C=64
    const float SC_B  = 0.57735026918962576f;   // sqrt(1/3)
    const float SC_PT = 0.13608276348795434f;   // sqrt(1/54)
    float lg = SC_QK * dot + SC_B * bt
             - 0.5f * SC_PT * hw * pts
             + 100000.0f * (maskA[(size_t)n * KK + k] - 1.0f);
    logit[k * 4 + h] = lg;
    __syncthreads();

    if (tid < 4) {
        float mx = -1e30f;
        for (int k2 = 0; k2 < KK; ++k2) mx = fmaxf(mx, logit[k2 * 4 + tid]);
        red[tid] = mx;
    }
    __syncthreads();
    float ev = expf(lg - red[h]);
    aw[k * 4 + h] = ev;
    __syncthreads();
    if (tid < 4) {
        float sm = 0.f;
        for (int k2 = 0; k2 < KK; ++k2) sm += aw[k2 * 4 + tid];
        red[4 + tid] = 1.0f / sm;
    }
    __syncthreads();
    aw[k * 4 + h] = ev * red[4 + h];
    __syncthreads();

    float* srow = S + (size_t)n * SDIM;

    // o: channels tid and tid+128 of H*C=256
    {
        int ch0 = tid, ch1 = tid + 128;
        int h0 = ch0 >> 6, c0 = ch0 & 63;
        int h1 = ch1 >> 6, c1 = ch1 & 63;
        float a0 = 0.f, a1 = 0.f;
        for (int k2 = 0; k2 < KK; ++k2) {
            const float* vrow = Ylin + (size_t)eix[k2] * NPROJ + 256 + 64;
            a0 += aw[k2 * 4 + h0] * vrow[h0 * 128 + c0];
            a1 += aw[k2 * 4 + h1] * vrow[h1 * 128 + c1];
        }
        srow[ch0] = a0;
        srow[ch1] = a1;
    }
    // o_pair: 64 channels, 4 heads each
    if (tid < DD) {
        float a0 = 0.f, a1 = 0.f, a2 = 0.f, a3 = 0.f;
        for (int k2 = 0; k2 < KK; ++k2) {
            float val = hE[((size_t)n * KK + k2) * DD + tid];
            a0 += aw[k2 * 4 + 0] * val;
            a1 += aw[k2 * 4 + 1] * val;
            a2 += aw[k2 * 4 + 2] * val;
            a3 += aw[k2 * 4 + 3] * val;
        }
        srow[384 +   0 + tid] = a0;
        srow[384 +  64 + tid] = a1;
        srow[384 + 128 + tid] = a2;
        srow[384 + 192 + tid] = a3;
    }
    // o_pt raw accumulation (96 = H*VP*3)
    if (tid < 96) {
        int hh = tid / 24, rem = tid % 24;
        float acc = 0.f;
        for (int k2 = 0; k2 < KK; ++k2) {
            float val = kvpts[(size_t)eix[k2] * 144 + hh * 36 + 12 + rem];
            acc += aw[k2 * 4 + hh] * val;
        }
        opt_raw[tid] = acc;
    }
    __syncthreads();
    // inverse frame + norm (32 = H*VP points)
    if (tid < 32) {
        int base = (tid >> 3) * 24 + (tid & 7) * 3;
        float ox = opt_raw[base + 0] - Rt_s[9];
        float oy = opt_raw[base + 1] - Rt_s[10];
        float oz = opt_raw[base + 2] - Rt_s[11];
        // out_i = sum_j R[j][i] * o_j   (R^T apply)
        float r0 = Rt_s[0]*ox + Rt_s[3]*oy + Rt_s[6]*oz;
        float r1 = Rt_s[1]*ox + Rt_s[4]*oy + Rt_s[7]*oz;
        float r2 = Rt_s[2]*ox + Rt_s[5]*oy + Rt_s[8]*oz;
        float nr = sqrtf(r0*r0 + r1*r1 + r2*r2 + 1e-8f);
        srow[256 + tid] = r0;
        srow[288 + tid] = r1;
        srow[320 + tid] = r2;
        srow[352 + tid] = nr;
    }
}

// ---------------------------------------------------------------------------
// out = LN(X + Yd) * (optional mask)   (row dim = 64)
// ---------------------------------------------------------------------------
__global__ void add_ln_kernel(const float* __restrict__ X, const float* __restrict__ Yd,
                              const float* __restrict__ g, const float* __restrict__ bta,
                              const float* __restrict__ maskV, float* __restrict__ out)
{
    const int row = blockIdx.x;
    const int t   = threadIdx.x;   // 64
    __shared__ float red[64];
    float v = X[(size_t)row * DD + t] + Yd[(size_t)row * DD + t];
    red[t] = v;
    __syncthreads();
    for (int s = 32; s > 0; s >>= 1) { if (t < s) red[t] += red[t + s]; __syncthreads(); }
    float mu = red[0] * (1.0f / 64.0f);
    __syncthreads();
    float d = v - mu;
    red[t] = d * d;
    __syncthreads();
    for (int s = 32; s > 0; s >>= 1) { if (t < s) red[t] += red[t + s]; __syncthreads(); }
    float var = red[0] * (1.0f / 64.0f);
    float o = d * rsqrtf(var + 1e-5f) * g[t] + bta[t];
    if (maskV) o *= maskV[row];
    out[(size_t)row * DD + t] = o;
}

// ---------------------------------------------------------------------------
extern "C" void kernel_launch(void* const* d_in, const int* in_sizes, int n_in,
                              void* d_out, int out_size, void* d_ws, size_t ws_size,
                              hipStream_t stream)
{
    const float* h_V    = (const float*)d_in[0];
    const float* h_E    = (const float*)d_in[1];
    const int*   E_idx  = (const int*)  d_in[2];
    const float* X      = (const float*)d_in[3];
    const float* mask_V = (const float*)d_in[4];
    const float* mask_A = (const float*)d_in[5];
    const float* Wq     = (const float*)d_in[6];
    const float* bq     = (const float*)d_in[7];
    const float* Wkv    = (const float*)d_in[8];
    const float* bkv    = (const float*)d_in[9];
    const float* Wqp    = (const float*)d_in[10];
    const float* bqp    = (const float*)d_in[11];
    const float* Wkvp   = (const float*)d_in[12];
    const float* bkvp   = (const float*)d_in[13];
    const float* Wb     = (const float*)d_in[14];
    const float* bb     = (const float*)d_in[15];
    const float* hwts   = (const float*)d_in[16];
    const float* Wout   = (const float*)d_in[17];
    const float* bout   = (const float*)d_in[18];
    const float* ln0g   = (const float*)d_in[19];
    const float* ln0b   = (const float*)d_in[20];
    const float* ln1g   = (const float*)d_in[21];
    const float* ln1b   = (const float*)d_in[22];
    const float* Wd1    = (const float*)d_in[23];
    const float* bd1    = (const float*)d_in[24];
    const float* Wd2    = (const float*)d_in[25];
    const float* bd2    = (const float*)d_in[26];
    float* out = (float*)d_out;

    // workspace layout (floats)
    float* ws    = (float*)d_ws;
    float* Wcat  = ws;                       size_t off = (size_t)DD * NPROJ;
    float* bcat  = ws + off;                 off += NPROJ;
    float* Fr    = ws + off;                 off += (size_t)NN * 12;
    float* Ylin  = ws + off;                 off += (size_t)NN * NPROJ;
    float* qpts  = ws + off;                 off += (size_t)NN * 48;
    float* kvpts = ws + off;                 off += (size_t)NN * 144;
    float* Sbuf  = ws + off;                 off += (size_t)NN * SDIM;
    float* Ubuf  = ws + off;                 off += (size_t)NN * DD;
    float* H1    = ws + off;                 off += (size_t)NN * DD;
    float* M1    = ws + off;                 off += (size_t)NN * 4 * DD;
    float* M2    = ws + off;                 off += (size_t)NN * DD;

    // 1) pack weights
    pack_w_kernel<<<(DD * NPROJ + 255) / 256, 256, 0, stream>>>(
        Wq, bq, Wkv, bkv, Wqp, bqp, Wkvp, bkvp, Wcat, bcat);

    // 2) frames
    frames_kernel<<<(NN + 127) / 128, 128, 0, stream>>>(X, Fr);

    // 3) combined projections: Ylin = h_V @ Wcat + bcat   (4096 x 960 x 64)
    gemm_kernel<<<dim3(NN / 64, NPROJ / 64), 128, 0, stream>>>(
        h_V, Wcat, bcat, Ylin, NN, NPROJ, DD, 0);

    // 4) frame-transform points
    frame_points_kernel<<<(NN + 127) / 128, 128, 0, stream>>>(Ylin, Fr, qpts, kvpts);

    // 5) attention -> s
    attn_kernel<<<NN, 128, 0, stream>>>(Ylin, qpts, kvpts, Fr, h_E, E_idx,
                                        mask_A, Wb, bb, hwts, Sbuf);

    // 6) u = s @ Wout + bout   (4096 x 64 x 640)
    gemm_kernel<<<dim3(NN / 64, DD / 64), 128, 0, stream>>>(
        Sbuf, Wout, bout, Ubuf, NN, DD, SDIM, 0);

    // 7) h1 = LN(h_V + u)
    add_ln_kernel<<<NN, 64, 0, stream>>>(h_V, Ubuf, ln0g, ln0b, nullptr, H1);

    // 8) m1 = relu(h1 @ Wd1 + bd1)   (4096 x 256 x 64)
    gemm_kernel<<<dim3(NN / 64, (4 * DD) / 64), 128, 0, stream>>>(
        H1, Wd1, bd1, M1, NN, 4 * DD, DD, 1);

    // 9) m2 = m1 @ Wd2 + bd2   (4096 x 64 x 256)
    gemm_kernel<<<dim3(NN / 64, DD / 64), 128, 0, stream>>>(
        M1, Wd2, bd2, M2, NN, DD, 4 * DD, 0);

    // 10) out = LN(h1 + m2) * mask_V
    add_ln_kernel<<<NN, 64, 0, stream>>>(H1, M2, ln1g, ln1b, mask_V, out);

    (void)in_sizes; (void)n_in; (void)out_size; (void)ws_size; (void)off;
}